// BertCRF_32134945309078
// MI455X (gfx1250) — compile-verified
//
#include <hip/hip_runtime.h>

typedef float v2f __attribute__((ext_vector_type(2)));
typedef float v8f __attribute__((ext_vector_type(8)));

#define HDIM 768
#define LBL  9
#define SEQ  512
#define BATCH 64
#define NPAD 16

// -------- Kernel 1: feats[row][n] = sum_k hidden[row][k] * Wpad[k][n] + b[n]
// via exact-fp32 V_WMMA_F32_16X16X4_F32. One wave per 16-row tile, 8 waves/block.
__global__ __launch_bounds__(256) void crf_proj_wmma(
    const float* __restrict__ hidden, const float* __restrict__ W,
    const float* __restrict__ bias, float* __restrict__ feats)
{
    __shared__ float Wlds[HDIM * NPAD];   // 48 KB, zero-padded to N=16
    for (int idx = threadIdx.x; idx < HDIM * NPAD; idx += 256) {
        int k = idx >> 4, n = idx & 15;
        Wlds[idx] = (n < LBL) ? W[k * LBL + n] : 0.0f;
    }
    __syncthreads();

    const int lane     = threadIdx.x & 31;
    const int wave     = threadIdx.x >> 5;
    const int tileBase = blockIdx.x * 128 + wave * 16;
    const int mRow     = lane & 15;        // A-matrix row for this lane
    const int kOff     = (lane >> 4) * 2;  // lanes 16-31 hold K=+2,+3
    const int n        = lane & 15;        // B/C/D column for this lane

    const float* ap = hidden + (size_t)(tileBase + mRow) * HDIM + kOff;

    v8f acc = {};
    #pragma unroll 4
    for (int k0 = 0; k0 < HDIM; k0 += 4) {
        v2f a;
        a.x = ap[k0];
        a.y = ap[k0 + 1];
        v2f bm;
        bm.x = Wlds[(k0 + kOff) * NPAD + n];
        bm.y = Wlds[(k0 + kOff + 1) * NPAD + n];
        acc = __builtin_amdgcn_wmma_f32_16x16x4_f32(
            false, a, false, bm, (short)0, acc, false, false);
    }

    const float bn    = (n < LBL) ? bias[n] : 0.0f;
    const int   mHalf = (lane >> 4) * 8;   // D: lanes 16-31 hold M=8..15
    #pragma unroll
    for (int r = 0; r < 8; ++r) {
        int row = tileBase + mHalf + r;
        feats[(size_t)row * NPAD + n] = acc[r] + bn;
    }
}

// -------- Kernel 2: Viterbi forward + backtrack. One wave per batch element,
// lane = label (9 active). Backpointers as bytes in LDS.
__global__ __launch_bounds__(32) void crf_viterbi(
    const float* __restrict__ feats, const int* __restrict__ mask,
    const float* __restrict__ startT, const float* __restrict__ endT,
    const float* __restrict__ trans, float* __restrict__ outScores,
    int* __restrict__ outPaths)
{
    __shared__ unsigned char bp[SEQ * LBL];   // backpointers
    __shared__ int pathLds[SEQ];

    const int b    = blockIdx.x;
    const int lane = threadIdx.x;
    const int l    = (lane < LBL) ? lane : (LBL - 1);  // clamp keeps loads in-bounds

    // transition column for this next-label, kept in registers
    float tcol[LBL];
    #pragma unroll
    for (int i = 0; i < LBL; ++i) tcol[i] = trans[i * LBL + l];

    const float* fb = feats + (size_t)b * SEQ * NPAD;
    const int*   mb = mask + b * SEQ;

    float score = startT[l] + fb[l];   // init = start_transitions + feats[0]

    for (int s = 1; s < SEQ; ++s) {
        float best = -3.402823466e+38f;
        int bestIdx = 0;
        #pragma unroll
        for (int i = 0; i < LBL; ++i) {
            float si   = __shfl(score, i, 32);
            float cand = si + tcol[i];
            if (cand > best) { best = cand; bestIdx = i; }  // strict > => first argmax
        }
        float nxt = best + fb[s * NPAD + l];
        if (lane < LBL) bp[s * LBL + lane] = (unsigned char)bestIdx;
        score = mb[s] ? nxt : score;
    }

    // final scores + argmax across labels
    float fin = score + endT[l];
    float best = -3.402823466e+38f;
    int bestIdx = 0;
    #pragma unroll
    for (int i = 0; i < LBL; ++i) {
        float si = __shfl(fin, i, 32);
        if (si > best) { best = si; bestIdx = i; }
    }

    if (lane == 0) {
        outScores[b] = best;
        int cur = bestIdx;
        pathLds[SEQ - 1] = cur;
        for (int s = SEQ - 2; s >= 0; --s) {
            int p = bp[(s + 1) * LBL + cur];
            cur = mb[s + 1] ? p : cur;   // freeze past sequence end
            pathLds[s] = cur;
        }
    }
    __syncthreads();
    for (int s = lane; s < SEQ; s += 32)
        outPaths[b * SEQ + s] = pathLds[s];
}

extern "C" void kernel_launch(void* const* d_in, const int* in_sizes, int n_in,
                              void* d_out, int out_size, void* d_ws, size_t ws_size,
                              hipStream_t stream) {
    const float* hidden = (const float*)d_in[0];
    const int*   amask  = (const int*)d_in[1];
    const float* W      = (const float*)d_in[2];
    const float* bias   = (const float*)d_in[3];
    const float* startT = (const float*)d_in[4];
    const float* endT   = (const float*)d_in[5];
    const float* trans  = (const float*)d_in[6];

    float* feats     = (float*)d_ws;                    // [32768][16] padded
    float* outScores = (float*)d_out;                   // [64]
    int*   outPaths  = (int*)((float*)d_out + BATCH);   // [64][512]

    // 32768 rows / (8 waves * 16 rows) = 256 blocks
    hipLaunchKernelGGL(crf_proj_wmma, dim3(256), dim3(256), 0, stream,
                       hidden, W, bias, feats);
    hipLaunchKernelGGL(crf_viterbi, dim3(BATCH), dim3(32), 0, stream,
                       feats, amask, startT, endT, trans, outScores, outPaths);
}